// S2ConvNet_original_15960098472506
// MI455X (gfx1250) — compile-verified
//
#include <hip/hip_runtime.h>
#include <math.h>

// ---------------- constants (fixed problem sizes) ----------------
#define BB   4      // batch
#define N0   224    // 2*B_IN
#define NL1  28     // B_L1
#define NB1  56     // 2*B_L1
#define NF1  55     // 2*NL1-1
#define NL2  14     // B_L2
#define NB2  28
#define NF2  27
#define NSP_S2 784  // NL1^2
#define NS1  29260  // nspec_so3(28)
#define NS2  3654   // nspec_so3(14)
#define C1 32
#define C2 64
#define C3 128
#define C4 64
#define C5 40

#define PI_D 3.14159265358979323846

__host__ __device__ __forceinline__ int nso3_off(int l){ return l*(4*l*l-1)/3; }

// ---------------- WMMA types ----------------
typedef __attribute__((ext_vector_type(2))) float v2f;
typedef __attribute__((ext_vector_type(8))) float v8f;

// ---------------- generic batched strided complex GEMM ----------------
// D[m,n] = sum_k op(A)[m,k] * B[k,n], planar complex (separate re/im planes),
// two-level index splits: m = m1*M2+m2, k = k1*K2+k2, n = n1*N2+n2.
// epilogue flags: 32=relu, 64=bias  (operand modes are template params)
struct GemmP {
  const float *Ar, *Ai, *Br, *Bi;
  float *Dr, *Di;
  const float *bias;
  int M, N, K;
  int M2, K2, N2;
  int sAm1, sAm2, sAk1, sAk2;
  int sBk1, sBk2, sBn1, sBn2;
  int sDm1, sDm2, sDn1, sDn2;
  long batA, batB, batD;
  int flags, biasDiv, biasMod, biasAdd;
};

#define TM 64
#define TN 64
#define KT 16

union U8 { v8f v; float f[8]; };

template<bool CONJA, bool AIMAG, bool BIMAG, bool REALOUT>
__global__ __launch_bounds__(256) void k_cgemm(GemmP p){
  __shared__ float sAr[TM][KT+1];
  __shared__ float sAi[TM][KT+1];
  __shared__ float sBr[KT][TN+1];
  __shared__ float sBi[KT][TN+1];

  const int tid  = threadIdx.x;
  const int lane = tid & 31;
  const int wave = tid >> 5;
  const int wm = wave >> 1;          // 0..3 (row tile)
  const int wn = wave & 1;           // 0..1 (col tile)
  const int bm = blockIdx.y * TM;
  const int bn = blockIdx.x * TN;
  const long batch = (long)blockIdx.z;

  const float* Ar = p.Ar + batch * p.batA;
  const float* Ai = AIMAG ? (p.Ai + batch * p.batA) : nullptr;
  const float* Br = p.Br + batch * p.batB;
  const float* Bi = BIMAG ? (p.Bi + batch * p.batB) : nullptr;

  U8 acc[2][4];
#pragma unroll
  for (int s = 0; s < 2; ++s)
#pragma unroll
    for (int j = 0; j < 4; ++j)
#pragma unroll
      for (int e = 0; e < 8; ++e) acc[s][j].f[e] = 0.f;

  const int half = (lane >> 4);           // 0 or 1
  const int mrow = wm*16 + (lane & 15);
  const int koff = (half << 1);

  for (int k0 = 0; k0 < p.K; k0 += KT){
    __syncthreads();
    // ---- load A tile 64 x KT ----
#pragma unroll
    for (int e0 = 0; e0 < TM*KT; e0 += 256){
      int e = e0 + tid;
      int ml = e >> 4, kl = e & (KT-1);
      int m = bm + ml, k = k0 + kl;
      float vr = 0.f, vi = 0.f;
      if (m < p.M && k < p.K){
        int m1 = m, m2 = 0;
        if (p.M2 > 1){ m1 = m / p.M2; m2 = m - m1*p.M2; }
        int k1 = k, k2 = 0;
        if (p.K2 > 1){ k1 = k / p.K2; k2 = k - k1*p.K2; }
        long off = (long)m1*p.sAm1 + (long)m2*p.sAm2 + (long)k1*p.sAk1 + (long)k2*p.sAk2;
        vr = Ar[off];
        if (AIMAG) vi = Ai[off];
      }
      sAr[ml][kl] = vr;
      if (AIMAG) sAi[ml][kl] = vi;
    }
    // ---- load B tile KT x 64 ----
#pragma unroll
    for (int e0 = 0; e0 < KT*TN; e0 += 256){
      int e = e0 + tid;
      int kl = e >> 6, nl = e & 63;
      int k = k0 + kl, n = bn + nl;
      float vr = 0.f, vi = 0.f;
      if (k < p.K && n < p.N){
        int k1 = k, k2 = 0;
        if (p.K2 > 1){ k1 = k / p.K2; k2 = k - k1*p.K2; }
        int n1 = n, n2 = 0;
        if (p.N2 > 1){ n1 = n / p.N2; n2 = n - n1*p.N2; }
        long off = (long)k1*p.sBk1 + (long)k2*p.sBk2 + (long)n1*p.sBn1 + (long)n2*p.sBn2;
        vr = Br[off];
        if (BIMAG) vi = Bi[off];
      }
      sBr[kl][nl] = vr;
      if (BIMAG) sBi[kl][nl] = vi;
    }
    // CDNA5 speculative prefetch of the next K-chunk (global_prefetch_b8)
    if ((k0 + KT) < p.K && tid == 0){
      __builtin_prefetch(Ar + (long)(k0+KT)*p.sAk1, 0, 1);
      __builtin_prefetch(Br + (long)(k0+KT)*p.sBk1, 0, 1);
    }
    __syncthreads();

    // ---- compute: K in steps of 4 via V_WMMA_F32_16X16X4_F32 ----
    // Template params make this straight-line: up to 32 back-to-back WMMAs.
#pragma unroll
    for (int kk = 0; kk < KT; kk += 4){
      const int ak = kk + koff;
      v2f ar; ar.x = sAr[mrow][ak]; ar.y = sAr[mrow][ak+1];
      v2f ai; ai.x = 0.f; ai.y = 0.f;
      if (AIMAG){ ai.x = sAi[mrow][ak]; ai.y = sAi[mrow][ak+1]; }
#pragma unroll
      for (int sub = 0; sub < 2; ++sub){
        const int bcol = wn*32 + sub*16 + (lane & 15);
        v2f br; br.x = sBr[ak][bcol]; br.y = sBr[ak+1][bcol];
        v2f bi; bi.x = 0.f; bi.y = 0.f;
        if (BIMAG){ bi.x = sBi[ak][bcol]; bi.y = sBi[ak+1][bcol]; }
        acc[sub][0].v = __builtin_amdgcn_wmma_f32_16x16x4_f32(false, ar, false, br, (short)0, acc[sub][0].v, false, false);
        if (BIMAG)
          acc[sub][1].v = __builtin_amdgcn_wmma_f32_16x16x4_f32(false, ar, false, bi, (short)0, acc[sub][1].v, false, false);
        if (AIMAG)
          acc[sub][2].v = __builtin_amdgcn_wmma_f32_16x16x4_f32(false, ai, false, br, (short)0, acc[sub][2].v, false, false);
        if (AIMAG && BIMAG)
          acc[sub][3].v = __builtin_amdgcn_wmma_f32_16x16x4_f32(false, ai, false, bi, (short)0, acc[sub][3].v, false, false);
      }
    }
  }

  // ---- epilogue ----
  float biasVal = 0.f;
  if (p.flags & 64)
    biasVal = p.bias[(int)((batch / p.biasDiv + p.biasAdd) % p.biasMod)];
  float* Dr = p.Dr + batch * p.batD;
  float* Di = (p.Di != nullptr) ? (p.Di + batch * p.batD) : nullptr;
#pragma unroll
  for (int sub = 0; sub < 2; ++sub){
    int n = bn + wn*32 + sub*16 + (lane & 15);
    if (n >= p.N) continue;
    int n1 = n, n2 = 0;
    if (p.N2 > 1){ n1 = n / p.N2; n2 = n - n1*p.N2; }
    long nOff = (long)n1*p.sDn1 + (long)n2*p.sDn2;
#pragma unroll
    for (int r = 0; r < 8; ++r){
      int m = bm + wm*16 + (half << 3) + r;
      if (m >= p.M) continue;
      int m1 = m, m2 = 0;
      if (p.M2 > 1){ m1 = m / p.M2; m2 = m - m1*p.M2; }
      long off = (long)m1*p.sDm1 + (long)m2*p.sDm2 + nOff;
      float rr = acc[sub][0].f[r], ri = acc[sub][1].f[r];
      float ir = acc[sub][2].f[r], ii = acc[sub][3].f[r];
      float dre = CONJA ? (rr + ii) : (rr - ii);
      if (REALOUT){
        float v = dre + biasVal;
        if (p.flags & 32) v = v > 0.f ? v : 0.f;
        Dr[off] = v;
      } else {
        float dim = CONJA ? (ri - ir) : (ri + ir);
        Dr[off] = dre;
        Di[off] = dim;
      }
    }
  }
}

// ---------------- table kernels ----------------
__global__ void k_quad(float* w, int b){
  int j = blockIdx.x*blockDim.x + threadIdx.x;
  int n = 2*b;
  if (j >= n) return;
  double s = 0.0;
  for (int k = 0; k < b; ++k)
    s += sin(PI_D * (2.0*j+1.0) * (2.0*k+1.0) / (4.0*b)) / (2.0*k+1.0);
  double v = 2.0/b * sin(PI_D*(2.0*j+1.0)/(4.0*b)) * s;
  w[j] = (float)(v * 2.0 * PI_D / ((double)n*n));
}

__global__ void k_dft(float* oR, float* oI, int rows, int cols, int nGrid,
                      int center, float sign, int freqIsRow){
  int idx = blockIdx.x*blockDim.x + threadIdx.x;
  if (idx >= rows*cols) return;
  int r = idx / cols, c = idx - r*cols;
  int f = (freqIsRow ? r : c) - center;
  int g = freqIsRow ? c : r;
  double ang = (double)sign * 2.0 * PI_D * (double)f * (double)g / (double)nGrid;
  double sv, cv; sincos(ang, &sv, &cv);
  oR[idx] = (float)cv; oI[idx] = (float)sv;
}

__device__ double wigner_d(int j, int mp, int m, double beta){
  double cb = cos(0.5*beta), sb = sin(0.5*beta);
  int smin = m - mp; if (smin < 0) smin = 0;
  int smax = j + m;  if (j - mp < smax) smax = j - mp;
  if (smax < smin) return 0.0;
  double pref = 0.5*(lgamma((double)(j+mp+1)) + lgamma((double)(j-mp+1)) +
                     lgamma((double)(j+m +1)) + lgamma((double)(j-m +1)));
  double sum = 0.0;
  for (int s = smin; s <= smax; ++s){
    double lden = lgamma((double)(j+m-s+1)) + lgamma((double)(s+1)) +
                  lgamma((double)(mp-m+s+1)) + lgamma((double)(j-mp-s+1));
    double t = exp(pref - lden) * pow(cb, (double)(2*j+m-mp-2*s)) * pow(sb, (double)(mp-m+2*s));
    sum += ((mp - m + s) & 1) ? -t : t;
  }
  return sum;
}

__global__ void k_wig_s2(float* out, const float* wq, int nbeta, int nl){
  int idx = blockIdx.x*blockDim.x + threadIdx.x;
  int tot = nbeta * nl * nl;
  if (idx >= tot) return;
  int bi = idx / (nl*nl);
  int r  = idx - bi*(nl*nl);
  int l = 0; while ((l+1)*(l+1) <= r) ++l;
  int mi = r - l*l;
  double beta = (bi + 0.5) / (double)nbeta * PI_D;
  out[idx] = (float)(wigner_d(l, mi - l, 0, beta) * (double)wq[bi]);
}

__global__ void k_wig_so3(float* out, const float* wq, int nbeta, int nl, int nspec, int weighted){
  int idx = blockIdx.x*blockDim.x + threadIdx.x;
  int tot = nbeta * nspec;
  if (idx >= tot) return;
  int bi = idx / nspec;
  int r  = idx - bi*nspec;
  int l = 0; while (nso3_off(l+1) <= r) ++l;
  int rem = r - nso3_off(l);
  int L = 2*l+1;
  int mi = rem / L, ni = rem - mi*L;
  double beta = (bi + 0.5) / (double)nbeta * PI_D;
  double d = wigner_d(l, mi - l, ni - l, beta);
  out[idx] = (float)(d * (weighted ? (double)wq[bi] : (double)(2*l+1)));
}

// ---------------- small contraction kernels ----------------
// spec1[r,z] = sum_beta wigS2[beta,r] * Xf1[z,beta,col(l,m)]
__global__ void k_s2fft_beta(const float* wig, const float* XfR, const float* XfI,
                             float* sR, float* sI, int zTot){
  int idx = blockIdx.x*blockDim.x + threadIdx.x;
  int tot = NSP_S2 * zTot;
  if (idx >= tot) return;
  int r = idx / zTot, z = idx - r*zTot;
  int l = 0; while ((l+1)*(l+1) <= r) ++l;
  int mi = r - l*l;
  int col = (NF1/2) + (mi - l);
  float ar = 0.f, ai = 0.f;
  for (int b = 0; b < N0; ++b){
    float w = wig[(long)b*NSP_S2 + r];
    long o = ((long)z*N0 + b)*NF1 + col;
    ar += w * XfR[o]; ai += w * XfI[o];
  }
  sR[(long)r*zTot + z] = ar;
  sI[(long)r*zTot + z] = ai;
}

// out[r2, zb*32+o] = sum_i spec1[l^2+mi, zb*3+i] * conj(w1[l^2+ni, i, o])
__global__ void k_s2_pointwise(const float* sR, const float* sI, const float* w1,
                               float* oR, float* oI){
  long idx = (long)blockIdx.x*blockDim.x + threadIdx.x;
  long tot = (long)NS1 * (BB*C1);
  if (idx >= tot) return;
  int r2  = (int)(idx / (BB*C1));
  int col = (int)(idx - (long)r2*(BB*C1));
  int zb = col / C1, o = col - zb*C1;
  int l = 0; while (nso3_off(l+1) <= r2) ++l;
  int rem = r2 - nso3_off(l);
  int L = 2*l+1;
  int mi = rem / L, ni = rem - mi*L;
  int rx = l*l + mi, ry = l*l + ni;
  float ar = 0.f, ai = 0.f;
  for (int i = 0; i < 3; ++i){
    float xr = sR[(long)rx*12 + zb*3 + i];
    float xi = sI[(long)rx*12 + zb*3 + i];
    const float* wp = w1 + (((long)ry*3 + i)*C1 + o)*2;
    float yr = wp[0], yi = -wp[1];           // conj
    ar += xr*yr - xi*yi;
    ai += xr*yi + xi*yr;
  }
  oR[idx] = ar; oI[idx] = ai;
}

// F[(zl*nb+b)*nf*nf + jm*nf + jn] = sum_l wigF[b, r(l,jm,jn)] * spec[r, zBase+zl]
__global__ void k_Fbuild(const float* wig, const float* sR, const float* sI,
                         float* FR, float* FI, int zc, int zBase, int zTot,
                         int nb, int nf, int nl, int nspec){
  int idx = blockIdx.x*blockDim.x + threadIdx.x;
  int tot = zc*nb*nf*nf;
  if (idx >= tot) return;
  int jn = idx % nf;
  int t  = idx / nf;
  int jm = t % nf;  t /= nf;
  int b  = t % nb;
  int zl = t / nb;
  int c = nl - 1;
  int mm = jm - c, nn = jn - c;
  int am = mm < 0 ? -mm : mm, an = nn < 0 ? -nn : nn;
  int lmin = am > an ? am : an;
  float ar = 0.f, ai = 0.f;
  for (int l = lmin; l < nl; ++l){
    int L = 2*l+1;
    int r = nso3_off(l) + (l+mm)*L + (l+nn);
    float w = wig[(long)b*nspec + r];
    long so = (long)r*zTot + zBase + zl;
    ar += w * sR[so]; ai += w * sI[so];
  }
  FR[idx] = ar; FI[idx] = ai;
}

// specx[r,z] = sum_beta wig[beta,r] * Xf[(z*nbin+beta)*NF2*NF2 + jm*NF2 + jn]
__global__ void k_so3fft_beta(const float* wig, const float* XfR, const float* XfI,
                              float* sR, float* sI, int zTot, int nbin){
  long idx = (long)blockIdx.x*blockDim.x + threadIdx.x;
  long tot = (long)NS2 * zTot;
  if (idx >= tot) return;
  int r = (int)(idx / zTot);
  int z = (int)(idx - (long)r*zTot);
  int l = 0; while (nso3_off(l+1) <= r) ++l;
  int rem = r - nso3_off(l);
  int L = 2*l+1;
  int mi = rem / L, ni = rem - mi*L;
  int jm = (NL2-1) + (mi - l), jn = (NL2-1) + (ni - l);
  float ar = 0.f, ai = 0.f;
  for (int b = 0; b < nbin; ++b){
    float w = wig[(long)b*NS2 + r];
    long o = (((long)z*nbin + b)*NF2 + jm)*NF2 + jn;
    ar += w * XfR[o]; ai += w * XfI[o];
  }
  sR[(long)r*zTot + z] = ar;
  sI[(long)r*zTot + z] = ai;
}

// so3_integrate: one block per (batch,channel)
__global__ void k_integrate(const float* h, const float* wq, float* feat){
  __shared__ float red[256];
  int bc = blockIdx.x;
  const float* base = h + (long)bc * (NB2*NB2*NB2);
  float s = 0.f;
  for (int i = threadIdx.x; i < NB2*NB2*NB2; i += 256)
    s += base[i] * wq[i / (NB2*NB2)];
  red[threadIdx.x] = s;
  __syncthreads();
  for (int st = 128; st > 0; st >>= 1){
    if (threadIdx.x < st) red[threadIdx.x] += red[threadIdx.x + st];
    __syncthreads();
  }
  if (threadIdx.x == 0) feat[bc] = red[0];
}

__global__ void k_head(const float* feat, const float* wout, const float* bout, float* out){
  int idx = threadIdx.x;
  if (idx >= BB*10) return;
  int z = idx / 10, j = idx - z*10;
  float s = bout[j];
  for (int c = 0; c < C5; ++c) s += feat[z*C5 + c] * wout[j*C5 + c];
  out[idx] = s;
}

// ---------------- host helpers ----------------
static inline GemmP gp0(){
  GemmP p{};
  p.M2 = p.K2 = p.N2 = 1;
  p.biasDiv = 1; p.biasMod = 1; p.biasAdd = 0;
  return p;
}

// flags bit0=conjA, bit1=Aimag, bit2=Bimag, bit3=realOut (+32 relu, +64 bias)
static inline void launch_gemm(hipStream_t s, const GemmP& p, int batches){
  dim3 g((unsigned)((p.N + TN - 1)/TN), (unsigned)((p.M + TM - 1)/TM), (unsigned)batches);
  dim3 b(256);
  const bool cA = (p.flags & 1);
  const bool aI = (p.flags & 2) && (p.Ai != nullptr);
  const bool bI = (p.flags & 4) && (p.Bi != nullptr);
  const bool rO = (p.flags & 8);
  if      (!cA && !aI &&  bI && !rO) k_cgemm<false,false,true ,false><<<g,b,0,s>>>(p);
  else if (!cA &&  aI &&  bI && !rO) k_cgemm<false,true ,true ,false><<<g,b,0,s>>>(p);
  else if ( cA &&  aI &&  bI && !rO) k_cgemm<true ,true ,true ,false><<<g,b,0,s>>>(p);
  else if (!cA &&  aI &&  bI &&  rO) k_cgemm<false,true ,true ,true ><<<g,b,0,s>>>(p);
  else                               k_cgemm<false,false,false,false><<<g,b,0,s>>>(p);
}

// forward SO3 transform: h real [z,beta,alpha,gamma] -> specx [NS2, zTot] (planar)
static void so3fwd(hipStream_t s, const float* h, int zTot, int nbin,
                   const float* EfR, const float* EfI, const float* wigW,
                   float* XpR, float* XpI, float* XfR, float* XfI,
                   float* sxR, float* sxI){
  int Q = zTot * nbin;
  {
    GemmP p = gp0();
    p.Ar = h; p.M = nbin; p.K = nbin; p.N = NF2;
    p.sAm1 = nbin; p.sAk1 = 1; p.batA = (long)nbin*nbin;
    p.Br = EfR; p.Bi = EfI; p.sBk1 = NF2; p.sBn1 = 1; p.batB = 0;
    p.Dr = XpR; p.Di = XpI; p.sDm1 = 1; p.sDn1 = nbin; p.batD = (long)NF2*nbin;
    p.flags = 4;
    launch_gemm(s, p, Q);
  }
  {
    GemmP p = gp0();
    p.Ar = XpR; p.Ai = XpI; p.M = NF2; p.K = nbin; p.N = NF2;
    p.sAm1 = nbin; p.sAk1 = 1; p.batA = (long)NF2*nbin;
    p.Br = EfR; p.Bi = EfI; p.sBk1 = NF2; p.sBn1 = 1; p.batB = 0;
    p.Dr = XfR; p.Di = XfI; p.sDm1 = 1; p.sDn1 = NF2; p.batD = (long)NF2*NF2;
    p.flags = 2 | 4;
    launch_gemm(s, p, Q);
  }
  long tot = (long)NS2 * zTot;
  k_so3fft_beta<<<dim3((unsigned)((tot + 255)/256)), dim3(256), 0, s>>>(wigW, XfR, XfI, sxR, sxI, zTot, nbin);
}

// per-l complex channel mixing: specy = conj(specx) x W  (the FLOP-dominant stage)
static void pointwise(hipStream_t s, const float* sxR, const float* sxI, int zIn,
                      const float* w, int Ci, int Co, float* syR, float* syI){
  int zOut = BB * Co;
  for (int l = 0; l < NL2; ++l){
    int L = 2*l + 1;
    long off = nso3_off(l);
    GemmP p = gp0();
    p.Ar = sxR + off*zIn; p.Ai = sxI + off*zIn;
    p.M = L*BB; p.M2 = BB; p.sAm1 = L*zIn; p.sAm2 = Ci;
    p.K = L*Ci; p.K2 = Ci; p.sAk1 = zIn;   p.sAk2 = 1;
    p.Br = w + off*(long)Ci*Co*2; p.Bi = p.Br + 1;
    p.N = L*Co; p.N2 = Co;
    p.sBn1 = L*Ci*Co*2; p.sBn2 = 2;
    p.sBk1 = Ci*Co*2;   p.sBk2 = Co*2;
    p.Dr = syR + off*zOut; p.Di = syI + off*zOut;
    p.sDm1 = L*zOut; p.sDm2 = Co; p.sDn1 = zOut; p.sDn2 = 1;
    p.flags = 1 | 2 | 4;   // conjA, complex A, complex B
    launch_gemm(s, p, 1);
  }
}

// inverse SO3 transform: spec [nspec, zTot] -> h real [z,beta,a2,a3] + bias + relu
static void rifft(hipStream_t s, const float* syR, const float* syI, int zTot,
                  const float* wigF, int nb, int nf, int nl, int nspec,
                  const float* EiR, const float* EiI,
                  float* FR, float* FI, float* PR, float* PI,
                  float* h, const float* bias, int Co, int zChunk){
  for (int z0 = 0; z0 < zTot; z0 += zChunk){
    int zc = zTot - z0 < zChunk ? zTot - z0 : zChunk;
    int tot = zc*nb*nf*nf;
    k_Fbuild<<<dim3((unsigned)((tot + 255)/256)), dim3(256), 0, s>>>(
        wigF, syR, syI, FR, FI, zc, z0, zTot, nb, nf, nl, nspec);
    {
      GemmP p = gp0();
      p.Ar = FR; p.Ai = FI; p.M = nf; p.K = nf; p.N = nb;
      p.sAm1 = 1; p.sAk1 = nf; p.batA = (long)nf*nf;
      p.Br = EiR; p.Bi = EiI; p.sBk1 = nb; p.sBn1 = 1; p.batB = 0;
      p.Dr = PR; p.Di = PI; p.sDm1 = nb; p.sDn1 = 1; p.batD = (long)nf*nb;
      p.flags = 2 | 4;
      launch_gemm(s, p, zc*nb);
    }
    {
      GemmP p = gp0();
      p.Ar = PR; p.Ai = PI; p.M = nb; p.K = nf; p.N = nb;
      p.sAm1 = 1; p.sAk1 = nb; p.batA = (long)nf*nb;
      p.Br = EiR; p.Bi = EiI; p.sBk1 = nb; p.sBn1 = 1; p.batB = 0;
      p.Dr = h + (long)z0*nb*nb*nb; p.sDm1 = nb; p.sDn1 = 1; p.batD = (long)nb*nb;
      p.flags = 2 | 4 | 8 | 32 | (bias ? 64 : 0);
      p.bias = bias; p.biasDiv = nb; p.biasMod = Co; p.biasAdd = z0;
      launch_gemm(s, p, zc*nb);
    }
  }
}

// ---------------- entry ----------------
extern "C" void kernel_launch(void* const* d_in, const int* in_sizes, int n_in,
                              void* d_out, int out_size, void* d_ws, size_t ws_size,
                              hipStream_t stream) {
  (void)in_sizes; (void)n_in; (void)out_size; (void)ws_size;
  const float* x   = (const float*)d_in[0];
  const float* w1  = (const float*)d_in[1];
  const float* bb1 = (const float*)d_in[2];
  const float* w2  = (const float*)d_in[3];
  const float* bb2 = (const float*)d_in[4];
  const float* w3  = (const float*)d_in[5];
  const float* bb3 = (const float*)d_in[6];
  const float* w4  = (const float*)d_in[7];
  const float* bb4 = (const float*)d_in[8];
  const float* w5  = (const float*)d_in[9];
  const float* bb5 = (const float*)d_in[10];
  const float* wout = (const float*)d_in[11];
  const float* bout = (const float*)d_in[12];
  float* out = (float*)d_out;

  char* ws = (char*)d_ws;
  size_t cur = 0;
  auto alloc = [&](size_t nFloats)->float*{
    float* p = (float*)(ws + cur);
    cur = (cur + nFloats*sizeof(float) + 255) & ~(size_t)255;
    return p;
  };

  // tables
  float* w112 = alloc(224);
  float* w28q = alloc(56);
  float* w14q = alloc(28);
  float* D224r = alloc(224*55); float* D224i = alloc(224*55);
  float* E56fr = alloc(56*27);  float* E56fi = alloc(56*27);
  float* E28fr = alloc(28*27);  float* E28fi = alloc(28*27);
  float* E56ir = alloc(55*56);  float* E56ii = alloc(55*56);
  float* E28ir = alloc(27*28);  float* E28ii = alloc(27*28);
  float* wigS2   = alloc((size_t)224*NSP_S2);
  float* wigF28  = alloc((size_t)56*NS1);
  float* wigW2814= alloc((size_t)56*NS2);
  float* wigW14  = alloc((size_t)28*NS2);
  float* wigF14  = alloc((size_t)28*NS2);
  // activations
  float* Xf1r = alloc((size_t)12*N0*NF1); float* Xf1i = alloc((size_t)12*N0*NF1);
  float* s1r  = alloc((size_t)NSP_S2*12); float* s1i  = alloc((size_t)NSP_S2*12);
  float* s2r  = alloc((size_t)NS1*(BB*C1)); float* s2i = alloc((size_t)NS1*(BB*C1));
  float* Fr = alloc(5420800); float* Fi = alloc(5420800);
  float* Pr = alloc(5519360); float* Pi = alloc(5519360);
  float* hA = alloc((size_t)(BB*C1)*NB1*NB1*NB1);   // h1 / h3 / h5
  float* hB = alloc((size_t)(BB*C2)*NB2*NB2*NB2);   // h2 / h4
  float* Xpr = alloc(10838016); float* Xpi = alloc(10838016);
  float* Xfr = alloc(10450944); float* Xfi = alloc(10450944);
  float* sxr = alloc((size_t)NS2*512); float* sxi = alloc((size_t)NS2*512);
  float* syr = alloc((size_t)NS2*512); float* syi = alloc((size_t)NS2*512);
  float* feat = alloc(BB*C5);

  auto g1 = [](long n){ return dim3((unsigned)((n + 255)/256)); };

  // ---- precompute tables (deterministic, every call) ----
  k_quad<<<g1(224), 256, 0, stream>>>(w112, 112);
  k_quad<<<g1(56),  256, 0, stream>>>(w28q, 28);
  k_quad<<<g1(28),  256, 0, stream>>>(w14q, 14);
  k_dft<<<g1(224*55), 256, 0, stream>>>(D224r, D224i, 224, 55, 224, 27, -1.f, 0);
  k_dft<<<g1(56*27),  256, 0, stream>>>(E56fr, E56fi, 56, 27, 56, 13, -1.f, 0);
  k_dft<<<g1(28*27),  256, 0, stream>>>(E28fr, E28fi, 28, 27, 28, 13, -1.f, 0);
  k_dft<<<g1(55*56),  256, 0, stream>>>(E56ir, E56ii, 55, 56, 56, 27,  1.f, 1);
  k_dft<<<g1(27*28),  256, 0, stream>>>(E28ir, E28ii, 27, 28, 28, 13,  1.f, 1);
  k_wig_s2 <<<g1((long)224*NSP_S2), 256, 0, stream>>>(wigS2, w112, 224, NL1);
  k_wig_so3<<<g1((long)56*NS1), 256, 0, stream>>>(wigF28,  nullptr, 56, NL1, NS1, 0);
  k_wig_so3<<<g1((long)56*NS2), 256, 0, stream>>>(wigW2814, w28q,   56, NL2, NS2, 1);
  k_wig_so3<<<g1((long)28*NS2), 256, 0, stream>>>(wigW14,   w14q,   28, NL2, NS2, 1);
  k_wig_so3<<<g1((long)28*NS2), 256, 0, stream>>>(wigF14,  nullptr, 28, NL2, NS2, 0);

  // ---- layer 1: s2_conv (b 112 -> 28) ----
  {
    // partial DFT over alpha: Xf1[(z*224+beta), col] = sum_a x * e^{-2pi i m a / 224}
    GemmP p = gp0();
    p.Ar = x; p.M = 12*N0; p.K = N0; p.N = NF1;
    p.sAm1 = N0; p.sAk1 = 1;
    p.Br = D224r; p.Bi = D224i; p.sBk1 = NF1; p.sBn1 = 1;
    p.Dr = Xf1r; p.Di = Xf1i; p.sDm1 = NF1; p.sDn1 = 1;
    p.flags = 4;
    launch_gemm(stream, p, 1);
  }
  k_s2fft_beta<<<g1((long)NSP_S2*12), 256, 0, stream>>>(wigS2, Xf1r, Xf1i, s1r, s1i, 12);
  k_s2_pointwise<<<g1((long)NS1*(BB*C1)), 256, 0, stream>>>(s1r, s1i, w1, s2r, s2i);
  rifft(stream, s2r, s2i, BB*C1, wigF28, NB1, NF1, NL1, NS1,
        E56ir, E56ii, Fr, Fi, Pr, Pi, hA, bb1, C1, 32);

  // ---- layer 2: so3_conv (b 28 -> 14), 32 -> 64 ----
  so3fwd(stream, hA, BB*C1, NB1, E56fr, E56fi, wigW2814, Xpr, Xpi, Xfr, Xfi, sxr, sxi);
  pointwise(stream, sxr, sxi, BB*C1, w2, C1, C2, syr, syi);
  rifft(stream, syr, syi, BB*C2, wigF14, NB2, NF2, NL2, NS2,
        E28ir, E28ii, Fr, Fi, Pr, Pi, hB, bb2, C2, 256);

  // ---- layer 3: 64 -> 128 ----
  so3fwd(stream, hB, BB*C2, NB2, E28fr, E28fi, wigW14, Xpr, Xpi, Xfr, Xfi, sxr, sxi);
  pointwise(stream, sxr, sxi, BB*C2, w3, C2, C3, syr, syi);
  rifft(stream, syr, syi, BB*C3, wigF14, NB2, NF2, NL2, NS2,
        E28ir, E28ii, Fr, Fi, Pr, Pi, hA, bb3, C3, 256);

  // ---- layer 4: 128 -> 64 ----
  so3fwd(stream, hA, BB*C3, NB2, E28fr, E28fi, wigW14, Xpr, Xpi, Xfr, Xfi, sxr, sxi);
  pointwise(stream, sxr, sxi, BB*C3, w4, C3, C4, syr, syi);
  rifft(stream, syr, syi, BB*C4, wigF14, NB2, NF2, NL2, NS2,
        E28ir, E28ii, Fr, Fi, Pr, Pi, hB, bb4, C4, 256);

  // ---- layer 5: 64 -> 40 ----
  so3fwd(stream, hB, BB*C4, NB2, E28fr, E28fi, wigW14, Xpr, Xpi, Xfr, Xfi, sxr, sxi);
  pointwise(stream, sxr, sxi, BB*C4, w5, C4, C5, syr, syi);
  rifft(stream, syr, syi, BB*C5, wigF14, NB2, NF2, NL2, NS2,
        E28ir, E28ii, Fr, Fi, Pr, Pi, hA, bb5, C5, 160);

  // ---- integrate + head ----
  k_integrate<<<dim3(BB*C5), dim3(256), 0, stream>>>(hA, w14q, feat);
  k_head<<<dim3(1), dim3(64), 0, stream>>>(feat, wout, bout, out);
}